// CornerNetPostProcessor_47270410060151
// MI455X (gfx1250) — compile-verified
//
#include <hip/hip_runtime.h>
#include <hip/hip_bf16.h>

// ---------------- problem constants (match setup_inputs) ----------------
#define B_    8
#define C_    80
#define H_    128
#define W_    128
#define HW_   (H_ * W_)          // 16384
#define CHW_  (C_ * HW_)         // 1,310,720
#define K_    100
#define NDET_ 1000
#define NSEL_ 16                 // 2 heats x 8 batches
#define NCHUNK_ 80               // CHW_ / 16384

typedef float v2f __attribute__((ext_vector_type(2)));
typedef float v8f __attribute__((ext_vector_type(8)));
typedef unsigned int  u32x4 __attribute__((ext_vector_type(4)));
typedef int           i32x4 __attribute__((ext_vector_type(4)));
typedef int           i32x8 __attribute__((ext_vector_type(8)));
typedef unsigned long long u64;
typedef unsigned int u32;

#if defined(__gfx1250__) && __has_builtin(__builtin_amdgcn_tensor_load_to_lds)
#define HAVE_TDM 1
#endif
#if __has_include(<hip/amd_detail/amd_gfx1250_TDM.h>)
#define TDM_6ARG 1               // therock-10.0 headers -> 6-arg builtin
#endif

// =====================================================================
// 1) fused sigmoid + 3x3 NMS. Row tile (10 x 128, halo rows) is staged
//    into LDS by the Tensor Data Mover (raw), then converted to sigmoid
//    with a -inf halo; falls back to plain loads if TDM is unavailable.
// =====================================================================
__global__ __launch_bounds__(256) void k_nms(const float* __restrict__ tl,
                                             const float* __restrict__ br,
                                             float* __restrict__ sup) {
  int id = blockIdx.x;
  int ytile = id % (H_ / 8); id /= (H_ / 8);
  int c     = id % C_;       id /= C_;
  int b     = id % B_;       id /= B_;
  int which = id;            // 0 = tl, 1 = br
  const float* src = (which == 0 ? tl : br) + ((size_t)b * C_ + c) * HW_;
  float* dst = sup + (size_t)which * (size_t)B_ * CHW_ + ((size_t)b * C_ + c) * (size_t)HW_;

  __shared__ float s[10][W_];          // sigmoid tile with halo
  int y0 = ytile * 8;

#ifdef HAVE_TDM
  __shared__ float rawT[10][W_];       // raw heat staging filled by TDM
  int ystart = (y0 - 1 < 0) ? 0 : y0 - 1;
  int yend   = (y0 + 9 > H_) ? H_ : y0 + 9;
  int nrows  = yend - ystart;
  int ly0    = ystart - (y0 - 1);      // 0 or 1
  if (threadIdx.x < 32) {              // TENSORcnt is per-wave: one wave issues
    u32 ldsOff = (u32)(uintptr_t)(&rawT[0][0]) + (u32)(ly0 * W_ * 4);
    unsigned long long ga = (unsigned long long)(uintptr_t)(src + (size_t)ystart * W_);
    // D# group 0: count=1 | lds_addr | global_addr[56:0] | type=2
    u32x4 g0;
    g0.x = 1u;
    g0.y = ldsOff;
    g0.z = (u32)(ga & 0xFFFFFFFFu);
    g0.w = (u32)((ga >> 32) & 0x01FFFFFFu) | (2u << 30);
    // D# group 1: data_size=4B, tensor_dim0=W, tensor_dim1=nrows,
    //             tile_dim0=W, tile_dim1=nrows, dim0_stride=W
    i32x8 g1;
    g1[0] = (int)(2u << 16);                                   // data_size=2 (4B)
    g1[1] = (int)(((u32)W_ & 0xFFFFu) << 16);                  // tensor_dim0 lo
    g1[2] = (int)(((u32)W_ >> 16) | (((u32)nrows & 0xFFFFu) << 16));
    g1[3] = (int)((((u32)nrows) >> 16) | (((u32)W_) << 16));   // tile_dim0
    g1[4] = nrows;                                             // tile_dim1
    g1[5] = W_;                                                // tensor_dim0_stride
    g1[6] = 0;
    g1[7] = 0;
    i32x4 gz = {0, 0, 0, 0};
#ifdef TDM_6ARG
    i32x8 gz8 = {0, 0, 0, 0, 0, 0, 0, 0};
    __builtin_amdgcn_tensor_load_to_lds(g0, g1, gz, gz, gz8, 0);
#else
    __builtin_amdgcn_tensor_load_to_lds(g0, g1, gz, gz, 0);
#endif
#if __has_builtin(__builtin_amdgcn_s_wait_tensorcnt)
    __builtin_amdgcn_s_wait_tensorcnt(0);
#endif
  }
  __syncthreads();                     // publish TDM-filled LDS to the block
  for (int i = threadIdx.x; i < 10 * W_; i += 256) {
    int ry = y0 - 1 + i / W_;
    float v;
    if (ry < 0 || ry >= H_) v = -__builtin_inff();
    else { float h = rawT[i / W_][i % W_]; v = 1.0f / (1.0f + __expf(-h)); }
    s[i / W_][i % W_] = v;
  }
#else
  for (int i = threadIdx.x; i < 10 * W_; i += 256) {
    int ry = y0 - 1 + i / W_;
    int rx = i % W_;
    float v;
    if (ry < 0 || ry >= H_) v = -__builtin_inff();
    else { float h = src[ry * W_ + rx]; v = 1.0f / (1.0f + __expf(-h)); }
    s[i / W_][rx] = v;
  }
#endif
  __syncthreads();
  // each thread produces 4 adjacent outputs -> one global_store_b128
  {
    int t  = threadIdx.x;              // 256 threads x 4 = 1024 = 8*128
    int ly = t >> 5;                   // 0..7
    int x0 = (t & 31) * 4;
    float o[4];
#pragma unroll
    for (int q = 0; q < 4; ++q) {
      int x = x0 + q;
      float cen = s[ly + 1][x];
      float m = cen;
#pragma unroll
      for (int dy = 0; dy < 3; ++dy) {
        float a  = (x > 0)      ? s[ly + dy][x - 1] : -__builtin_inff();
        float bv =                s[ly + dy][x];
        float cc = (x < W_ - 1) ? s[ly + dy][x + 1] : -__builtin_inff();
        m = fmaxf(m, fmaxf(a, fmaxf(bv, cc)));
      }
      o[q] = (m == cen) ? cen : 0.0f;
    }
    float4 v4 = make_float4(o[0], o[1], o[2], o[3]);
    *(float4*)(dst + (y0 + ly) * W_ + x0) = v4;
  }
}

// =====================================================================
// 2) top-K selection: 2-level radix select (all sup values are >= 0,
//    so raw IEEE bits are order-preserving)
// meta layout (u32): [0..15]=T1  [16..31]=cntAbove1  [32..47]=T2  [48..63]=candCount
// =====================================================================
__global__ void k_zero(u32* p, int n) {
  int i = blockIdx.x * 256 + threadIdx.x;
  if (i < n) p[i] = 0;
}

__global__ __launch_bounds__(256) void k_hist(const float* __restrict__ sup,
                                              u32* __restrict__ hist,
                                              const u32* __restrict__ meta,
                                              int pass) {
  int s = blockIdx.y;
  __shared__ u32 h[4096];
  for (int i = threadIdx.x; i < 4096; i += 256) h[i] = 0;
  __syncthreads();
  u32 T1 = meta[s];
  const float4* p4 = (const float4*)(sup + (size_t)s * CHW_ + (size_t)blockIdx.x * 16384);
  for (int i = threadIdx.x; i < 4096; i += 256) {     // 16k elems as float4
    __builtin_prefetch(p4 + i + 2048, 0, 1);          // global_prefetch
    float4 v = p4[i];
    u32 w[4] = {__float_as_uint(v.x), __float_as_uint(v.y),
                __float_as_uint(v.z), __float_as_uint(v.w)};
#pragma unroll
    for (int q = 0; q < 4; ++q) {
      if (pass == 0)                   atomicAdd(&h[w[q] >> 20], 1u);
      else if ((w[q] >> 20) == T1)     atomicAdd(&h[(w[q] >> 8) & 0xFFFu], 1u);
    }
  }
  __syncthreads();
  for (int i = threadIdx.x; i < 4096; i += 256)
    if (h[i]) atomicAdd(&hist[s * 4096 + i], h[i]);
}

__global__ __launch_bounds__(256) void k_scan(const u32* __restrict__ hist,
                                              u32* __restrict__ meta,
                                              int pass) {
  int s = blockIdx.x;
  int Kwant = (pass == 0) ? K_ : (int)(K_ - meta[16 + s]);
  __shared__ u32 ps[256];
  int t = threadIdx.x;
  u32 sum = 0;
#pragma unroll
  for (int i = 0; i < 16; ++i) sum += hist[s * 4096 + (4095 - (t * 16 + i))];
  u32 own = sum;
  ps[t] = sum; __syncthreads();
  for (int off = 1; off < 256; off <<= 1) {
    u32 v = (t >= off) ? ps[t - off] : 0; __syncthreads();
    ps[t] += v; __syncthreads();
  }
  u32 incl = ps[t], excl = incl - own;
  if ((int)excl < Kwant && Kwant <= (int)incl) {
    u32 acc = excl, T = 0, cnt = excl;
    for (int i = 0; i < 16; ++i) {
      int bin = 4095 - (t * 16 + i);
      u32 c = hist[s * 4096 + bin];
      if ((int)(acc + c) >= Kwant) { T = (u32)bin; cnt = acc; break; }
      acc += c;
    }
    if (pass == 0) { meta[s] = T; meta[16 + s] = cnt; }
    else           { meta[32 + s] = T; }
  }
}

__global__ __launch_bounds__(256) void k_compact(const float* __restrict__ sup,
                                                 const u32* __restrict__ meta,
                                                 u32* __restrict__ candCount,
                                                 u64* __restrict__ cand) {
  int s = blockIdx.y;
  u32 T1 = meta[s], T2 = meta[32 + s];
  const float4* p4 = (const float4*)(sup + (size_t)s * CHW_ + (size_t)blockIdx.x * 16384);
  u32 baseIdx = (u32)blockIdx.x * 16384u;
  for (int i = threadIdx.x; i < 4096; i += 256) {
    float4 v = p4[i];
    u32 w[4] = {__float_as_uint(v.x), __float_as_uint(v.y),
                __float_as_uint(v.z), __float_as_uint(v.w)};
#pragma unroll
    for (int q = 0; q < 4; ++q) {
      u32 hi = w[q] >> 20, mid = (w[q] >> 8) & 0xFFFu;
      if (hi > T1 || (hi == T1 && mid >= T2)) {
        u32 pos = atomicAdd(&candCount[s], 1u);
        if (pos < 1024)   // tie-break: lower flat index ranks first -> store ~idx
          cand[(size_t)s * 1024 + pos] =
              ((u64)w[q] << 32) | (u32)~(baseIdx + (u32)(i * 4 + q));
      }
    }
  }
}

__global__ __launch_bounds__(256) void k_top100(const u64* __restrict__ cand,
                                                const u32* __restrict__ meta,
                                                float* __restrict__ topS,
                                                int* __restrict__ topI) {
  int s = blockIdx.x;
  __shared__ u64 kv[1024];
  u32 n = meta[48 + s]; if (n > 1024) n = 1024;
  for (int i = threadIdx.x; i < 1024; i += 256)
    kv[i] = (i < (int)n) ? cand[(size_t)s * 1024 + i] : 0ull;
  __syncthreads();
  for (int k = 2; k <= 1024; k <<= 1)
    for (int j = k >> 1; j > 0; j >>= 1) {
      for (int i = threadIdx.x; i < 1024; i += 256) {
        int l = i ^ j;
        if (l > i) {
          u64 x = kv[i], y = kv[l];
          if ((x < y) == ((i & k) == 0)) { kv[i] = y; kv[l] = x; }  // descending
        }
      }
      __syncthreads();
    }
  for (int i = threadIdx.x; i < K_; i += 256) {
    u64 key = kv[i];
    topS[s * 128 + i] = __uint_as_float((u32)(key >> 32));
    topI[s * 128 + i] = (int)~(u32)(key & 0xFFFFFFFFu);
  }
}

// =====================================================================
// 3) pairwise KxK scoring (WMMA outer-product) + top-1000 via 16384-key
//    bitonic sort in 128KB LDS (CDNA5 320KB/WGP)
// =====================================================================
__global__ __launch_bounds__(1024) void k_pair(const float* __restrict__ tl_tag,
                                               const float* __restrict__ br_tag,
                                               const float* __restrict__ tl_regr,
                                               const float* __restrict__ br_regr,
                                               const float* __restrict__ topS,
                                               const int* __restrict__ topI,
                                               float* __restrict__ out) {
  int b = blockIdx.x;
  __shared__ u64 kv[16384];                    // 128 KB: packed (flipped score | ~idx)
  __shared__ float tlS[112], brS[112], tlT[112], brT[112];
  __shared__ float tlX[112], tlY[112], brX[112], brY[112];
  __shared__ int   tlC[112], brC[112];
  int tid = threadIdx.x;
  for (int i = tid; i < 16384; i += 1024) kv[i] = 0ull;

  if (tid < 112) {                             // gather TL side
    int t = tid;
    if (t < K_) {
      int sel = b;
      int ind = topI[sel * 128 + t];
      int cls = ind / HW_, rem = ind % HW_;
      tlS[t] = topS[sel * 128 + t];
      tlC[t] = cls;
      tlT[t] = tl_tag[(size_t)b * HW_ + rem];
      tlX[t] = (float)(rem % W_) + tl_regr[((size_t)b * 2)     * HW_ + rem];
      tlY[t] = (float)(rem / W_) + tl_regr[((size_t)b * 2 + 1) * HW_ + rem];
    } else { tlS[t] = 0.f; tlC[t] = -1; tlT[t] = 3.0e38f; tlX[t] = 0.f; tlY[t] = 0.f; }
  } else if (tid >= 128 && tid < 240) {        // gather BR side
    int t = tid - 128;
    if (t < K_) {
      int sel = 8 + b;
      int ind = topI[sel * 128 + t];
      int cls = ind / HW_, rem = ind % HW_;
      brS[t] = topS[sel * 128 + t];
      brC[t] = cls;
      brT[t] = br_tag[(size_t)b * HW_ + rem];
      brX[t] = (float)(rem % W_) + br_regr[((size_t)b * 2)     * HW_ + rem];
      brY[t] = (float)(rem / W_) + br_regr[((size_t)b * 2 + 1) * HW_ + rem];
    } else { brS[t] = 0.f; brC[t] = -2; brT[t] = -3.0e38f; brX[t] = 0.f; brY[t] = 0.f; }
  }
  __syncthreads();

  // (tl_i + br_j) * 0.5 is rank-2: A(16x4)=[0.5*a_i, 0.5, 0, 0], B(4x16)=[1; b_j; 0; 0]
  // One V_WMMA_F32_16X16X4_F32 per 16x16 tile; 7x7 = 49 tiles over 32 waves.
  // Operand build is branchless (lane mask multiply) so EXEC stays all-ones.
  int wave = tid >> 5, lane = tid & 31;
  float lmask = (lane < 16) ? 1.0f : 0.0f;
  for (int tile = wave; tile < 49; tile += 32) {
    int m0 = (tile / 7) * 16, n0 = (tile % 7) * 16;
    int lm = lane & 15;
    int n  = n0 + lm;
    int mB = m0 + ((lane >= 16) ? 8 : 0);
    v8f d;
#if __has_builtin(__builtin_amdgcn_wmma_f32_16x16x4_f32)
    // A layout (f32 16x4): lanes 0-15 M=lane {K0,K1}, lanes 16-31 M=lane-16 {K2,K3}
    v2f a, bm;
    a.x  = lmask * 0.5f * tlS[m0 + lm];
    a.y  = lmask * 0.5f;
    bm.x = lmask;
    bm.y = lmask * brS[n];
    v8f cacc = {};
    d = __builtin_amdgcn_wmma_f32_16x16x4_f32(false, a, false, bm, (short)0, cacc,
                                              false, false);
#else
#pragma unroll
    for (int r = 0; r < 8; ++r) d[r] = 0.5f * (tlS[mB + r] + brS[n]);
#endif
#pragma unroll
    for (int r = 0; r < 8; ++r) {
      int m = mB + r;
      if (m < K_ && n < K_) {
        float sc = d[r];
        bool invalid = (tlC[m] != brC[n]) | (fabsf(tlT[m] - brT[n]) > 0.5f)
                     | (brX[n] < tlX[m]) | (brY[n] < tlY[m]);
        if (invalid) sc = -1.0f;
        u32 fb = __float_as_uint(sc);
        fb ^= (fb & 0x80000000u) ? 0xFFFFFFFFu : 0x80000000u;   // orderable
        u32 sidx = (u32)(m * K_ + n);
        kv[sidx] = ((u64)fb << 32) | (u32)(16383u - sidx);      // low idx wins ties
      }
    }
  }
  __syncthreads();

  // full descending bitonic sort of 16384 u64 keys in LDS
  for (int k = 2; k <= 16384; k <<= 1) {
    for (int j = k >> 1; j > 0; j >>= 1) {
      for (int i = tid; i < 16384; i += 1024) {
        int l = i ^ j;
        if (l > i) {
          u64 x = kv[i], y = kv[l];
          if ((x < y) == ((i & k) == 0)) { kv[i] = y; kv[l] = x; }
        }
      }
      __syncthreads();
    }
  }

  // emit: bboxes [B,1000,4] | scores [B,1000] | clses | tl_sc | br_sc (all fp32)
  float* bbox = out;
  float* scor = out + (size_t)B_ * NDET_ * 4;
  float* clso = scor + (size_t)B_ * NDET_;
  float* tlo  = clso + (size_t)B_ * NDET_;
  float* bro  = tlo  + (size_t)B_ * NDET_;
  for (int i = tid; i < NDET_; i += 1024) {
    u64 key = kv[i];
    u32 fb = (u32)(key >> 32);
    fb ^= (fb & 0x80000000u) ? 0x80000000u : 0xFFFFFFFFu;       // un-flip
    float sc = __uint_as_float(fb);
    int sidx = (int)(16383u - (u32)(key & 0xFFFFFFFFu));
    int m = sidx / K_, n = sidx % K_;
    size_t o = (size_t)b * NDET_ + i;
    bbox[o * 4 + 0] = tlX[m]; bbox[o * 4 + 1] = tlY[m];
    bbox[o * 4 + 2] = brX[n]; bbox[o * 4 + 3] = brY[n];
    scor[o] = sc;
    clso[o] = (float)(tlC[m] + 1);
    tlo[o]  = tlS[m];
    bro[o]  = brS[n];
  }
}

// =====================================================================
// host launcher
// =====================================================================
extern "C" void kernel_launch(void* const* d_in, const int* in_sizes, int n_in,
                              void* d_out, int out_size, void* d_ws, size_t ws_size,
                              hipStream_t stream) {
  const float* tl_heat = (const float*)d_in[0];
  const float* br_heat = (const float*)d_in[1];
  const float* tl_tag  = (const float*)d_in[2];
  const float* br_tag  = (const float*)d_in[3];
  const float* tl_regr = (const float*)d_in[4];
  const float* br_regr = (const float*)d_in[5];
  float* out = (float*)d_out;

  // workspace layout
  char* ws = (char*)d_ws;
  size_t SUP_BYTES  = (size_t)2 * B_ * CHW_ * sizeof(float);     // 83.9 MB
  float* sup   = (float*)ws;
  u32*   hist1 = (u32*)(ws + SUP_BYTES);
  u32*   hist2 = hist1 + NSEL_ * 4096;
  u32*   meta  = hist2 + NSEL_ * 4096;                           // 64 u32
  u64*   cand  = (u64*)(meta + 64);                              // 16 x 1024
  float* topS  = (float*)(cand + NSEL_ * 1024);                  // 16 x 128
  int*   topI  = (int*)(topS + NSEL_ * 128);                     // 16 x 128

  // zero hist1 + hist2 + meta (contiguous)
  int nz = NSEL_ * 4096 * 2 + 64;
  k_zero<<<(nz + 255) / 256, 256, 0, stream>>>(hist1, nz);

  // fused sigmoid + NMS over both heatmaps (TDM-staged tiles)
  k_nms<<<2 * B_ * C_ * (H_ / 8), 256, 0, stream>>>(tl_heat, br_heat, sup);

  // two-level radix select -> top-100 per (heat,batch)
  dim3 hg(NCHUNK_, NSEL_);
  k_hist   <<<hg, 256, 0, stream>>>(sup, hist1, meta, 0);
  k_scan   <<<NSEL_, 256, 0, stream>>>(hist1, meta, 0);
  k_hist   <<<hg, 256, 0, stream>>>(sup, hist2, meta, 1);
  k_scan   <<<NSEL_, 256, 0, stream>>>(hist2, meta, 1);
  k_compact<<<hg, 256, 0, stream>>>(sup, meta, meta + 48, cand);
  k_top100 <<<NSEL_, 256, 0, stream>>>(cand, meta, topS, topI);

  // pairwise WMMA scoring + top-1000 + output emit
  k_pair<<<B_, 1024, 0, stream>>>(tl_tag, br_tag, tl_regr, br_regr, topS, topI, out);

  (void)in_sizes; (void)n_in; (void)out_size; (void)ws_size;
}